// RAA_51874615001249
// MI455X (gfx1250) — compile-verified
//
#include <hip/hip_runtime.h>
#include <hip/hip_bf16.h>

typedef __attribute__((ext_vector_type(2))) float v2f;
typedef __attribute__((ext_vector_type(8))) float v8f;

#define KDIM 8

// ---------------------------------------------------------------------------
// K1: per-column softmax of Zp (K=8, axis=0), store Z[n*8+k] row-major by n,
//     and deterministically accumulate per-block partial column sums of
//     ZtG[n][k] = Z[k][n] * sigmoid(Gp[n][k]).
// ---------------------------------------------------------------------------
__global__ void k1_softmax_colsum(const float* __restrict__ Zp,
                                  const float* __restrict__ Gp,
                                  float* __restrict__ Zout,
                                  float* __restrict__ cpart, int N) {
  __shared__ float sred[256][KDIM];
  int n = blockIdx.x * blockDim.x + threadIdx.x;
  float part[KDIM];
  if (n < N) {
    float zt[KDIM];
    float mx = -3.4e38f;
    for (int k = 0; k < KDIM; ++k) { zt[k] = Zp[(size_t)k * N + n]; mx = fmaxf(mx, zt[k]); }
    float sum = 0.f;
    for (int k = 0; k < KDIM; ++k) { zt[k] = expf(zt[k] - mx); sum += zt[k]; }
    float inv = 1.0f / sum;
    for (int k = 0; k < KDIM; ++k) {
      zt[k] *= inv;
      Zout[(size_t)n * KDIM + k] = zt[k];
      float g = 1.0f / (1.0f + expf(-Gp[(size_t)n * KDIM + k]));
      part[k] = zt[k] * g;
    }
  } else {
    for (int k = 0; k < KDIM; ++k) part[k] = 0.f;
  }
  for (int k = 0; k < KDIM; ++k) sred[threadIdx.x][k] = part[k];
  __syncthreads();
  for (int s = 128; s > 0; s >>= 1) {
    if (threadIdx.x < (unsigned)s)
      for (int k = 0; k < KDIM; ++k) sred[threadIdx.x][k] += sred[threadIdx.x + s][k];
    __syncthreads();
  }
  if (threadIdx.x < KDIM) cpart[blockIdx.x * KDIM + threadIdx.x] = sred[0][threadIdx.x];
}

// ---------------------------------------------------------------------------
// K2: finalize colsum (fixed-order sum over K1 blocks) and sp_a = softplus(a)
// ---------------------------------------------------------------------------
__global__ void k2_finalize(const float* __restrict__ cpart, int nblocks,
                            const float* __restrict__ a,
                            float* __restrict__ colsum, float* __restrict__ spa) {
  int t = threadIdx.x;
  if (t < KDIM) {
    float s = 0.f;
    for (int b = 0; b < nblocks; ++b) s += cpart[b * KDIM + t];
    colsum[t] = s;
  }
  if (t == KDIM) {
    float av = a[0];
    spa[0] = (av > 20.f) ? av : log1pf(expf(av));
  }
}

// ---------------------------------------------------------------------------
// K3: A = Zs @ Cs  (8x8).  64 threads, one output each, fixed-order s loop.
//     Cs[s][q] = Z[n][q] * sigmoid(Gp[n][q]) / colsum[q],  n = sample_idx[s]
// ---------------------------------------------------------------------------
__global__ void k3_build_A(const float* __restrict__ Z, const float* __restrict__ Gp,
                           const int* __restrict__ sidx, const float* __restrict__ colsum,
                           float* __restrict__ Amat, int S) {
  int t = threadIdx.x;            // 64 threads
  int p = t >> 3, q = t & 7;
  float invc = 1.0f / colsum[q];
  float acc = 0.f;
  for (int s = 0; s < S; ++s) {
    int n = sidx[s];
    float zp = Z[(size_t)n * KDIM + p];
    float zq = Z[(size_t)n * KDIM + q];
    float g  = 1.0f / (1.0f + expf(-Gp[(size_t)n * KDIM + q]));
    acc += zp * zq * g;
  }
  Amat[t] = acc * invc;
}

// ---------------------------------------------------------------------------
// K4: M[s][p] = sum_q A[p][q] * Z[sample_idx[s]][q]; also ||m_s||^2, row sums,
//     and beta_s.
// ---------------------------------------------------------------------------
__global__ void k4_build_M(const float* __restrict__ Z, const float* __restrict__ Amat,
                           const int* __restrict__ sidx, const float* __restrict__ beta,
                           float* __restrict__ M, float* __restrict__ nm,
                           float* __restrict__ srow, float* __restrict__ bs, int S) {
  __shared__ float sA[64];
  if (threadIdx.x < 64) sA[threadIdx.x] = Amat[threadIdx.x];
  __syncthreads();
  int s = blockIdx.x * blockDim.x + threadIdx.x;
  if (s >= S) return;
  int n = sidx[s];
  float z[KDIM];
  for (int q = 0; q < KDIM; ++q) z[q] = Z[(size_t)n * KDIM + q];
  float nrm = 0.f, sr = 0.f;
  for (int p = 0; p < KDIM; ++p) {
    float m = 0.f;
    for (int q = 0; q < KDIM; ++q) m += sA[p * KDIM + q] * z[q];
    M[(size_t)s * KDIM + p] = m;
    nrm += m * m;
    sr += m;
  }
  nm[s] = nrm; srow[s] = sr; bs[s] = beta[n];
}

// ---------------------------------------------------------------------------
// K5: WMMA Gram tiles (fp32 16x16x4, K=8 -> two chained WMMAs) + distance
//     epilogue + exp accumulation.
//       dist^2 = nm[i] + nm[j] - 2*dot + 2e-6*(srow[i]-srow[j]) + 8e-12
//
//     Wave mapping: total waves = nt*16.  Each wave owns one row-tile ti and
//     an 8-tile column chunk, so the A-fragments and all i-side statistics
//     (nm/srow/bs for this lane's 8 rows) are loaded ONCE and reused across
//     the inner tj sweep.  Inner loop per tile: 4 B-frag loads + 3 j-side
//     gathers + 2 v_wmma_f32_16x16x4_f32 + exp epilogue.
//
//     Fragment layouts (ISA 7.12.2, 32-bit):
//       A (16x4): lane<16 -> row r0, VGPR0/1 = K0/K1; lane>=16 -> K2/K3.
//       B (4x16) mirrors it, so A- and B-side per-lane gathers are identical.
//       C/D: VGPR r, lane<16 -> (M=r, N=lane); lane>=16 -> (M=r+8, N=lane-16).
// ---------------------------------------------------------------------------
__global__ void k5_pdist1(const float* __restrict__ M, const float* __restrict__ nm,
                          const float* __restrict__ srow, const float* __restrict__ bs,
                          const float* __restrict__ spa_p, float* __restrict__ part1,
                          int nt /* = S/16 */) {
  __shared__ float red[256];
  const float spa = spa_p[0];
  int lane = threadIdx.x & 31;
  int w = (int)((blockIdx.x * blockDim.x + threadIdx.x) >> 5);  // global wave id
  int r0 = lane & 15;   // row within tile for A/B frags; also column for C/D
  int kh = lane >> 4;   // k-half select
  float acc = 0.f;

  const int cpr = 16;                 // column chunks per row-tile
  int tpc = (nt + cpr - 1) / cpr;     // tiles per chunk (nt=128 -> 8)
  int ti    = w / cpr;                // row-tile owned by this wave
  int chunk = w % cpr;

  if (ti < nt) {
    int i0 = ti * 16;
    // ---- hoisted A-side: fragments + i-row statistics for this lane ----
    const float* Mi = M + (size_t)(i0 + r0) * KDIM + 2 * kh;
    v2f a0 = {Mi[0], Mi[1]};   // K = 0..3 half
    v2f a1 = {Mi[4], Mi[5]};   // K = 4..7 half
    float nmi[8], sri[8], bi[8];
    for (int r = 0; r < 8; ++r) {
      int i = i0 + r + 8 * kh;
      nmi[r] = nm[i]; sri[r] = srow[i]; bi[r] = bs[i];
    }
    int tj_end = chunk * tpc + tpc;
    if (tj_end > nt) tj_end = nt;
    for (int tj = chunk * tpc; tj < tj_end; ++tj) {
      int j0 = tj * 16;
      const float* Mj = M + (size_t)(j0 + r0) * KDIM + 2 * kh;
      v2f b0 = {Mj[0], Mj[1]};
      v2f b1 = {Mj[4], Mj[5]};
      v8f c = {};
      c = __builtin_amdgcn_wmma_f32_16x16x4_f32(false, a0, false, b0, (short)0, c, false, false);
      c = __builtin_amdgcn_wmma_f32_16x16x4_f32(false, a1, false, b1, (short)0, c, false, false);
      int j = j0 + r0;
      float nmj = nm[j], sj = srow[j], bj = bs[j];
      for (int r = 0; r < 8; ++r) {
        int i = i0 + r + 8 * kh;
        if (i == j) continue;                  // mat_nd removes diagonal
        float d2 = nmi[r] + nmj - 2.f * c[r] + 2e-6f * (sri[r] - sj) + 8e-12f;
        d2 = fmaxf(d2, 0.f);
        acc += expf(bi[r] + bj - spa * sqrtf(d2));
      }
    }
  }

  red[threadIdx.x] = acc;
  __syncthreads();
  for (int s = 128; s > 0; s >>= 1) {
    if (threadIdx.x < (unsigned)s) red[threadIdx.x] += red[threadIdx.x + s];
    __syncthreads();
  }
  if (threadIdx.x == 0) part1[blockIdx.x] = red[0];
}

// ---------------------------------------------------------------------------
// K6: sparse part.  per edge: v_k = 1e-6 + sum_q A[k][q]*(z_i[q]-z_j[q]);
//     contrib = beta[i]+beta[j] - spa*||v||.
// ---------------------------------------------------------------------------
__global__ void k6_pdist2(const float* __restrict__ Z, const float* __restrict__ Amat,
                          const float* __restrict__ beta, const int* __restrict__ si,
                          const int* __restrict__ sj, const float* __restrict__ spa_p,
                          float* __restrict__ part2, int E) {
  __shared__ float sA[64];
  __shared__ float red[256];
  if (threadIdx.x < 64) sA[threadIdx.x] = Amat[threadIdx.x];
  __syncthreads();
  const float spa = spa_p[0];
  float acc = 0.f;
  int stride = (int)(gridDim.x * blockDim.x);
  for (int e = (int)(blockIdx.x * blockDim.x + threadIdx.x); e < E; e += stride) {
    int ni = si[e], nj = sj[e];
    float w[KDIM];
    for (int q = 0; q < KDIM; ++q)
      w[q] = Z[(size_t)ni * KDIM + q] - Z[(size_t)nj * KDIM + q];
    float ss = 0.f;
    for (int k = 0; k < KDIM; ++k) {
      float v = 1e-6f;
      for (int q = 0; q < KDIM; ++q) v += sA[k * KDIM + q] * w[q];
      ss += v * v;
    }
    acc += beta[ni] + beta[nj] - spa * sqrtf(ss);
  }
  red[threadIdx.x] = acc;
  __syncthreads();
  for (int s = 128; s > 0; s >>= 1) {
    if (threadIdx.x < (unsigned)s) red[threadIdx.x] += red[threadIdx.x + s];
    __syncthreads();
  }
  if (threadIdx.x == 0) part2[blockIdx.x] = red[0];
}

// ---------------------------------------------------------------------------
// K7: fixed-order final combine: out = z_pdist2 - 0.5*e^2*sum(exp terms)
// ---------------------------------------------------------------------------
__global__ void k7_final(const float* __restrict__ part1, int n1,
                         const float* __restrict__ part2, int n2,
                         float* __restrict__ out) {
  if (threadIdx.x == 0 && blockIdx.x == 0) {
    float s1 = 0.f, s2 = 0.f;
    for (int i = 0; i < n1; ++i) s1 += part1[i];
    for (int i = 0; i < n2; ++i) s2 += part2[i];
    float e = expf(1.0f);
    out[0] = s2 - 0.5f * e * e * s1;
  }
}

extern "C" void kernel_launch(void* const* d_in, const int* in_sizes, int n_in,
                              void* d_out, int out_size, void* d_ws, size_t ws_size,
                              hipStream_t stream) {
  (void)n_in; (void)out_size; (void)ws_size;
  const float* beta = (const float*)d_in[0];
  const float* a    = (const float*)d_in[1];
  const float* Zp   = (const float*)d_in[2];
  const float* Gp   = (const float*)d_in[3];
  const int* sidx   = (const int*)d_in[4];
  const int* spi    = (const int*)d_in[5];
  const int* spj    = (const int*)d_in[6];
  float* out = (float*)d_out;

  const int N = in_sizes[0];   // 100000
  const int S = in_sizes[4];   // 2048
  const int E = in_sizes[5];   // 131072
  const int nB1 = (N + 255) / 256;
  const int nt  = S / 16;                    // 128 row/col tiles
  const int GB5 = (nt * 16 + 7) / 8;         // waves = nt*16, 8 waves/block -> 256
  const int GB6 = 256;

  // workspace layout (floats)
  float* ws    = (float*)d_ws;
  float* Zbuf  = ws;                                  // N*8
  float* cpart = Zbuf + (size_t)N * KDIM;             // nB1*8
  float* colsum = cpart + (size_t)nB1 * KDIM;         // 8
  float* spa   = colsum + KDIM;                       // 1 (+7 pad)
  float* Amat  = spa + 8;                             // 64
  float* Mbuf  = Amat + 64;                           // S*8
  float* nm    = Mbuf + (size_t)S * KDIM;             // S
  float* srow  = nm + S;                              // S
  float* bs    = srow + S;                            // S
  float* part1 = bs + S;                              // GB5
  float* part2 = part1 + GB5;                         // GB6

  k1_softmax_colsum<<<nB1, 256, 0, stream>>>(Zp, Gp, Zbuf, cpart, N);
  k2_finalize<<<1, 64, 0, stream>>>(cpart, nB1, a, colsum, spa);
  k3_build_A<<<1, 64, 0, stream>>>(Zbuf, Gp, sidx, colsum, Amat, S);
  k4_build_M<<<(S + 255) / 256, 256, 0, stream>>>(Zbuf, Amat, sidx, beta, Mbuf, nm, srow, bs, S);
  k5_pdist1<<<GB5, 256, 0, stream>>>(Mbuf, nm, srow, bs, spa, part1, nt);
  k6_pdist2<<<GB6, 256, 0, stream>>>(Zbuf, Amat, beta, spi, spj, spa, part2, E);
  k7_final<<<1, 32, 0, stream>>>(part1, GB5, part2, GB6, out);
}